// TopKActivation_67284957659715
// MI455X (gfx1250) — compile-verified
//
#include <hip/hip_runtime.h>

// TopKActivation: per row of x[8192][16384], keep top-k (k=128) values, zero rest.
// Memory-bound: 1 GB total traffic -> ~45us floor at 23.3 TB/s on MI455X.
// Strategy: 1 block (8 wave32s) per row, row lives in VGPRs, exact k-th value
// found with a 4-round 8-bit MSD radix select using wave-privatized LDS
// histograms; single NT read + single NT write of the row.

typedef __attribute__((ext_vector_type(4))) float v4f;

constexpr int D_DIM  = 16384;          // row length (fixed by reference)
constexpr int TPB    = 256;            // 8 waves of 32 on CDNA5
constexpr int VPT    = D_DIM / TPB;    // 64 values per thread (in VGPRs)
constexpr int NWAVES = TPB / 32;       // wave32

// Order-preserving float -> uint mapping (monotone: a<b  <=>  f2ord(a)<f2ord(b))
__device__ __forceinline__ unsigned f2ord(float f) {
  unsigned b = __float_as_uint(f);
  return (b & 0x80000000u) ? ~b : (b | 0x80000000u);
}
__device__ __forceinline__ float ord2f(unsigned u) {
  unsigned b = (u & 0x80000000u) ? (u ^ 0x80000000u) : ~u;
  return __uint_as_float(b);
}

__global__ __launch_bounds__(TPB)
void topk_act_kernel(const float* __restrict__ x,
                     const int* __restrict__ kptr,
                     float* __restrict__ out) {
  __shared__ unsigned hist[NWAVES][256];   // wave-privatized histograms (8 KB)
  __shared__ unsigned s_prefix;
  __shared__ unsigned s_kr;
  __shared__ unsigned s_ctr;

  const int tid = threadIdx.x;
  const long long row = blockIdx.x;
  const float* xr   = x   + row * (long long)D_DIM;
  float*       outr = out + row * (long long)D_DIM;
  const unsigned k = (unsigned)(*kptr);

  // ---- Load the row once, coalesced NT b128, convert to sortable uints ----
  const v4f* xr4 = reinterpret_cast<const v4f*>(xr);
  unsigned u[VPT];
#pragma unroll
  for (int c = 0; c < VPT / 4; ++c) {
    v4f v = __builtin_nontemporal_load(xr4 + (c * TPB + tid));
    u[c * 4 + 0] = f2ord(v.x);
    u[c * 4 + 1] = f2ord(v.y);
    u[c * 4 + 2] = f2ord(v.z);
    u[c * 4 + 3] = f2ord(v.w);
  }

  v4f* o4 = reinterpret_cast<v4f*>(outr);

  if (k >= (unsigned)D_DIM) {            // degenerate: identity
#pragma unroll
    for (int c = 0; c < VPT / 4; ++c) {
      v4f o;
      o.x = ord2f(u[c * 4 + 0]);
      o.y = ord2f(u[c * 4 + 1]);
      o.z = ord2f(u[c * 4 + 2]);
      o.w = ord2f(u[c * 4 + 3]);
      __builtin_nontemporal_store(o, o4 + (c * TPB + tid));
    }
    return;
  }

  const int w = tid >> 5;                // wave id (wave32 on gfx1250)
  unsigned prefix = 0;                   // known high bits of the k-th value
  unsigned kr = k;                       // ranks still to resolve among matches

  // ---- 4-round MSD radix select over the 32-bit sortable encoding ----
  for (int r = 0; r < 4; ++r) {
    const int shift = 24 - 8 * r;

    // clear privatized histograms (2048 words / 256 threads)
    for (int i = tid; i < NWAVES * 256; i += TPB)
      (&hist[0][0])[i] = 0u;
    __syncthreads();

    const unsigned hi_mask = (r == 0) ? 0u : (0xFFFFFFFFu << (shift + 8));
#pragma unroll
    for (int j = 0; j < VPT; ++j) {
      unsigned v = u[j];
      if ((v & hi_mask) == prefix)
        atomicAdd(&hist[w][(v >> shift) & 255u], 1u);
    }
    __syncthreads();

    // merge wave-private histograms: thread t owns bin t (no cross-thread deps)
    unsigned binc = 0;
#pragma unroll
    for (int i = 0; i < NWAVES; ++i) binc += hist[i][tid];
    hist[0][tid] = binc;
    __syncthreads();

    // scan from the top digit down to locate the digit holding rank kr
    if (tid == 0) {
      unsigned cum = 0;
      for (int d = 255; d >= 0; --d) {
        unsigned c = hist[0][d];
        if (cum + c >= kr) {
          s_prefix = prefix | ((unsigned)d << shift);
          s_kr     = kr - cum;
          break;
        }
        cum += c;
      }
    }
    __syncthreads();
    prefix = s_prefix;
    kr     = s_kr;
  }

  if (tid == 0) s_ctr = 0u;
  __syncthreads();

  // ---- Emit: keep values > T; keep first kr values == T; zero the rest ----
  const unsigned T = prefix;
#pragma unroll
  for (int c = 0; c < VPT / 4; ++c) {
    float of[4];
#pragma unroll
    for (int j = 0; j < 4; ++j) {
      unsigned v = u[c * 4 + j];
      float f = 0.0f;
      if (v > T) {
        f = ord2f(v);
      } else if (v == T) {
        unsigned pos = atomicAdd(&s_ctr, 1u);
        if (pos < kr) f = ord2f(v);
      }
      of[j] = f;
    }
    v4f o;
    o.x = of[0]; o.y = of[1]; o.z = of[2]; o.w = of[3];
    __builtin_nontemporal_store(o, o4 + (c * TPB + tid));
  }
}

extern "C" void kernel_launch(void* const* d_in, const int* in_sizes, int n_in,
                              void* d_out, int out_size, void* d_ws, size_t ws_size,
                              hipStream_t stream) {
  const float* x    = (const float*)d_in[0];
  const int*   kptr = (const int*)d_in[1];
  float*       out  = (float*)d_out;

  const int B = in_sizes[0] / D_DIM;     // 8192 rows
  topk_act_kernel<<<B, TPB, 0, stream>>>(x, kptr, out);
}